// CompProbModel_76948634075343
// MI455X (gfx1250) — compile-verified
//
#include <hip/hip_runtime.h>
#include <math.h>

// Collapsed TTI completion-probability model for MI455X (gfx1250, wave32).
//
// The reference's (B,6600,40,22) tensor is reduced by three gathers to 8
// scalars, so the optimal kernel evaluates only the selected field cell /
// time index per batch: one wave32 per batch, lane j = player j (22 active,
// 10 neutral lanes), wave-wide product reduction over defenders via lane
// permutes, then a lane broadcast to pick the receiver's probability.

#define J_PLAYERS 22
#define NFEAT     14
#define A_MAX_C   7.25f
#define S_MAX_C   9.25f
#define SIG_K     (3.14f / (1.732f * 0.5f))   // 3.14 / (1.732 * TTI_SIGMA)

__global__ __launch_bounds__(32)
void tti_collapsed_kernel(const float* __restrict__ frame,
                          float* __restrict__ out, int B) {
    const int b = blockIdx.x;
    if (b >= B) return;
    const int j = threadIdx.x;                 // lane = player index
    const float* fb = frame + (size_t)b * J_PLAYERS * NFEAT;

    // Batch-level broadcast features live in every row; read from row 0.
    const float bx   = fb[11];                 // ball end x (integer-valued)
    const float by   = fb[12];                 // ball end y (integer-valued)
    const float toff = fb[13];                 // time-of-flight (integer-valued)

    const int   ix = (int)bx;                  // ball_end_x
    const int   iy = (int)by + 1;              // ball_end_y + 1 (row index)
    const float lx = (float)ix + 0.5f;         // x = linspace(0.5,119.5,120)[ix]
    const float ly = (iy == 0) ? -0.2f : ((float)iy - 0.5f);  // y[.0]=-0.2 special case
    const int   tk = (int)rintf(toff) - 1;     // selected T index
    const float Tt = 0.1f + (3.9f / 39.0f) * (float)tk;       // linspace(0.1,4.0,40)[tk]

    float p      = 0.0f;   // interception prob of this player at (cell, Tt)
    float factor = 1.0f;   // (1 - p * (1-team)) : neutral 1 for attackers / idle lanes
    int   pm_cand = 0;     // receiver-lane candidate (one-hot -> exactly one lane != 0-ish)

    if (j < J_PLAYERS) {
        const float* fj = fb + j * NFEAT;
        const float px = fj[1],  py = fj[2];
        const float vx = fj[3],  vy = fj[4];
        const float team = fj[7];
        const float rec  = fj[10];

        // REAX_T == 0: positions/velocities are used as-is.
        const float dx = lx - px;
        const float dy = ly - py;
        const float dmag = sqrtf(dx * dx + dy * dy);

        float s0 = (dx * vx + dy * vy) / dmag;
        s0 = fminf(fmaxf(s0, -S_MAX_C), S_MAX_C);

        float t_lt = (S_MAX_C - s0) / A_MAX_C;
        const float d_lt0 = t_lt * (s0 + S_MAX_C) * 0.5f;
        if (d_lt0 > dmag) {
            const float q = s0 / A_MAX_C;
            t_lt = -q + sqrtf(q * q + 2.0f * dmag / A_MAX_C);
        }
        const float d_lt = fminf(fmaxf(d_lt0, 0.0f), dmag);
        const float t_at = (dmag - d_lt) / S_MAX_C;
        const float t_tot = t_lt + t_at;       // + REAX_T (= 0)

        const float z = SIG_K * (Tt - t_tot);  // + TTI_EPS (= 0)
        p = 1.0f / (1.0f + expf(-z));

        factor  = 1.0f - p * (1.0f - team);    // defenders: 1-p ; attackers: 1
        pm_cand = (rec > 0.5f) ? j : 0;        // argmax(rec*idx) == one-hot index
    }

    // Wave32 reductions (lane permutes -> gfx1250 ds/permlane path).
    float prod_def = factor;
    int   pm       = pm_cand;
    #pragma unroll
    for (int m = 16; m >= 1; m >>= 1) {
        prod_def *= __shfl_xor(prod_def, m, 32);
        const int o = __shfl_xor(pm, m, 32);
        pm = (o > pm) ? o : pm;
    }
    const float p_rec = __shfl(p, pm, 32);     // broadcast receiver's p

    if (j == 0) {
        // team[pm] == 1 always (tgt in [0,11)); lambda_off == 1 -> identity power.
        out[b] = p_rec * prod_def + 0.001f;
    }
}

extern "C" void kernel_launch(void* const* d_in, const int* in_sizes, int n_in,
                              void* d_out, int out_size, void* d_ws, size_t ws_size,
                              hipStream_t stream) {
    (void)n_in; (void)d_ws; (void)ws_size;
    const float* frame = (const float*)d_in[0];
    float* out = (float*)d_out;
    const int B = in_sizes[0] / (J_PLAYERS * NFEAT);   // = 8 = out_size
    (void)out_size;
    tti_collapsed_kernel<<<dim3(B), dim3(32), 0, stream>>>(frame, out, B);
}